// SAN_59459527246207
// MI455X (gfx1250) — compile-verified
//
#include <hip/hip_runtime.h>
#include <hip/hip_bf16.h>
#include <math.h>

// ---------------------------------------------------------------------------
// Problem constants (from reference): B=4, n=512, in_dim=300, h=128
// ---------------------------------------------------------------------------
#define BATCH   4
#define SEQ     512
#define INDIM   300
#define HDIM    128
#define MROWS   (BATCH * SEQ)       // 2048 rows for all GEMMs

typedef __attribute__((ext_vector_type(2))) float v2f;
typedef __attribute__((ext_vector_type(8))) float v8f;

// ---------------------------------------------------------------------------
// WMMA fragment loaders for V_WMMA_F32_16X16X4_F32 (wave32).
// A (16x4, MxK): lanes 0-15 hold M=lane, VGPR0=K0/K2 VGPR1=K1/K3 split at lane16.
//   -> lane L holds A[M = L&15][k0 + 2*(L>>4) + {0,1}]
// B (4x16, KxN): row striped across lanes within a VGPR, K split at lane 16
//   (mirrors the C/D half-split): lane L holds B[k0 + 2*(L>>4) + {0,1}][N = L&15]
// C/D (16x16): VGPR r -> M = r + 8*(lane>>4), N = lane&15.
// ---------------------------------------------------------------------------

// A row-major [rows x lda]; returns {A[m][ka], A[m][ka+1]} per the layout above.
__device__ __forceinline__ v2f load_fragA(const float* A, int m0, int k0,
                                          int lda, int lane) {
    const int m  = m0 + (lane & 15);
    const int ka = k0 + ((lane >> 4) << 1);
    return *reinterpret_cast<const v2f*>(A + (size_t)m * lda + ka);   // 8B aligned (lda even, ka even)
}

// We need B[k][n] = W[n][k] (W row-major [HDIM x ldw]); since W rows are
// contiguous in k, lane L loads W[n0 + (L&15)][kb], W[...][kb+1].
__device__ __forceinline__ v2f load_fragB_WT(const float* W, int n0, int k0,
                                             int ldw, int lane) {
    const int nn = n0 + (lane & 15);
    const int kb = k0 + ((lane >> 4) << 1);
    return *reinterpret_cast<const v2f*>(W + (size_t)nn * ldw + kb);
}

// ---------------------------------------------------------------------------
// Kernel 1: h_vecs = tanh(word_vecs @ Wx^T + bx)   [2048 x 300]x[300 x 128]
// 8 waves / block, one 16x16 output tile per wave, K-loop 75 steps of 4.
// ---------------------------------------------------------------------------
__global__ __launch_bounds__(256) void gemm_wx_tanh(
    const float* __restrict__ A,      // word_vecs [2048 x 300]
    const float* __restrict__ W,      // Wx_w      [128 x 300]
    const float* __restrict__ bias,   // Wx_b      [128]
    float* __restrict__ out)          // h_vecs    [2048 x 128]
{
    const int lane = threadIdx.x & 31;
    const int wid  = blockIdx.x * 8 + (threadIdx.x >> 5);   // 1024 tiles total
    const int m0   = (wid >> 3) << 4;                       // 128 M-tiles
    const int n0   = (wid & 7) << 4;                        // 8 N-tiles

    v8f acc = {};
#pragma unroll 5
    for (int k0 = 0; k0 < INDIM; k0 += 4) {
        v2f a = load_fragA(A, m0, k0, INDIM, lane);
        v2f b = load_fragB_WT(W, n0, k0, INDIM, lane);
        acc = __builtin_amdgcn_wmma_f32_16x16x4_f32(
                false, a, false, b, (short)0, acc, false, false);
    }

    const int col = n0 + (lane & 15);
    const int rb  = m0 + ((lane >> 4) << 3);
    const float bv = bias[col];
#pragma unroll
    for (int r = 0; r < 8; ++r) {
        out[(size_t)(rb + r) * HDIM + col] = tanhf(acc[r] + bv);
    }
}

// ---------------------------------------------------------------------------
// Kernel 2: ta = h_vecs @ Wa^T + ba ; tb = h_vecs @ Ua^T + bu (fused: one A
// fragment feeds two WMMA accumulators). K = 128 -> 32 steps of 4.
// ---------------------------------------------------------------------------
__global__ __launch_bounds__(256) void gemm_ta_tb(
    const float* __restrict__ A,      // h_vecs [2048 x 128]
    const float* __restrict__ Wa,     // [128 x 128]
    const float* __restrict__ ba,
    const float* __restrict__ Ua,     // [128 x 128]
    const float* __restrict__ bu,
    float* __restrict__ ta,           // [2048 x 128]
    float* __restrict__ tb)           // [2048 x 128]
{
    const int lane = threadIdx.x & 31;
    const int wid  = blockIdx.x * 8 + (threadIdx.x >> 5);
    const int m0   = (wid >> 3) << 4;
    const int n0   = (wid & 7) << 4;

    v8f accA = {};
    v8f accB = {};
#pragma unroll 4
    for (int k0 = 0; k0 < HDIM; k0 += 4) {
        v2f a  = load_fragA(A, m0, k0, HDIM, lane);
        v2f bw = load_fragB_WT(Wa, n0, k0, HDIM, lane);
        v2f bu2 = load_fragB_WT(Ua, n0, k0, HDIM, lane);
        accA = __builtin_amdgcn_wmma_f32_16x16x4_f32(
                 false, a, false, bw, (short)0, accA, false, false);
        accB = __builtin_amdgcn_wmma_f32_16x16x4_f32(
                 false, a, false, bu2, (short)0, accB, false, false);
    }

    const int col = n0 + (lane & 15);
    const int rb  = m0 + ((lane >> 4) << 3);
    const float bva = ba[col];
    const float bvu = bu[col];
#pragma unroll
    for (int r = 0; r < 8; ++r) {
        const size_t idx = (size_t)(rb + r) * HDIM + col;
        ta[idx] = accA[r] + bva;
        tb[idx] = accB[r] + bvu;
    }
}

// ---------------------------------------------------------------------------
// Kernel 3: fused per-channel attention with online softmax over k.
//   score = elu(ta[b,q,h] + tb[b,k,h]) + mask[b,k,h]
//   out[b,q,h] = sum_k softmax_k(score) * h_vecs[b,k,h]
// Block = 128 threads (thread = channel h, fully coalesced), QT=4 q-streams
// per thread amortize every tb/hv/mask load 4x. Data lives in L2 (192MB).
// ---------------------------------------------------------------------------
#define QT 4

__global__ __launch_bounds__(128) void attn_fused(
    const float* __restrict__ ta,     // [B*SEQ x H]
    const float* __restrict__ tb,     // [B*SEQ x H]
    const float* __restrict__ hv,     // [B*SEQ x H]
    const float* __restrict__ maskp,  // [B*SEQ x H]
    float* __restrict__ out)          // [B*SEQ x H]
{
    const int h   = threadIdx.x;                 // 0..127
    const int bq  = blockIdx.x;                  // B * SEQ/QT blocks
    const int b   = bq / (SEQ / QT);
    const int q0  = (bq % (SEQ / QT)) * QT;
    const size_t base = (size_t)b * SEQ * HDIM;

    float taq[QT], m[QT], s[QT], acc[QT];
#pragma unroll
    for (int i = 0; i < QT; ++i) {
        taq[i] = ta[base + (size_t)(q0 + i) * HDIM + h];
        m[i]   = -INFINITY;
        s[i]   = 0.0f;
        acc[i] = 0.0f;
    }

    const float* ptb = tb    + base + h;
    const float* phv = hv    + base + h;
    const float* pmk = maskp + base + h;

    for (int k = 0; k < SEQ; ++k) {
        // prefetch a few rows ahead (lowers to global_prefetch_b8 on gfx1250)
        if (k + 16 < SEQ) {
            __builtin_prefetch(ptb + (size_t)(k + 16) * HDIM, 0, 0);
            __builtin_prefetch(phv + (size_t)(k + 16) * HDIM, 0, 0);
        }
        const float tbk = ptb[(size_t)k * HDIM];
        const float hvk = phv[(size_t)k * HDIM];
        const float mkk = pmk[(size_t)k * HDIM];
#pragma unroll
        for (int i = 0; i < QT; ++i) {
            const float x = taq[i] + tbk;
            const float e = (x > 0.0f ? x : __expf(x) - 1.0f) + mkk;  // elu + mask
            const float mn   = fmaxf(m[i], e);
            const float corr = __expf(m[i] - mn);   // 0 on first iter (m=-inf)
            const float p    = __expf(e - mn);
            s[i]   = s[i]  * corr + p;
            acc[i] = acc[i] * corr + p * hvk;
            m[i]   = mn;
        }
    }

#pragma unroll
    for (int i = 0; i < QT; ++i) {
        out[base + (size_t)(q0 + i) * HDIM + h] = acc[i] / s[i];
    }
}

// ---------------------------------------------------------------------------
// Host-side launcher
// Inputs: 0 word_vecs, 1 mask, 2 Wx_w, 3 Wx_b, 4 Wa_w, 5 Wa_b, 6 Ua_w, 7 Ua_b
// Workspace layout: hv (1MB) | ta (1MB) | tb (1MB)
// ---------------------------------------------------------------------------
extern "C" void kernel_launch(void* const* d_in, const int* in_sizes, int n_in,
                              void* d_out, int out_size, void* d_ws, size_t ws_size,
                              hipStream_t stream) {
    const float* word_vecs = (const float*)d_in[0];
    const float* mask      = (const float*)d_in[1];
    const float* Wx_w      = (const float*)d_in[2];
    const float* Wx_b      = (const float*)d_in[3];
    const float* Wa_w      = (const float*)d_in[4];
    const float* Wa_b      = (const float*)d_in[5];
    const float* Ua_w      = (const float*)d_in[6];
    const float* Ua_b      = (const float*)d_in[7];
    float* out = (float*)d_out;

    float* hv = (float*)d_ws;                       // [2048 x 128]
    float* ta = hv + (size_t)MROWS * HDIM;          // [2048 x 128]
    float* tb = ta + (size_t)MROWS * HDIM;          // [2048 x 128]

    // 1024 16x16 tiles, 8 waves (256 threads) per block -> 128 blocks
    gemm_wx_tanh<<<128, 256, 0, stream>>>(word_vecs, Wx_w, Wx_b, hv);
    gemm_ta_tb<<<128, 256, 0, stream>>>(hv, Wa_w, Wa_b, Ua_w, Ua_b, ta, tb);

    // B * SEQ/QT = 4 * 128 = 512 blocks of 128 threads
    attn_fused<<<BATCH * (SEQ / QT), 128, 0, stream>>>(ta, tb, hv, mask, out);
}